// ScatLayerj2_42949672961270
// MI455X (gfx1250) — compile-verified
//
#include <hip/hip_runtime.h>
#include <math.h>

typedef __attribute__((ext_vector_type(2))) float v2f;
typedef __attribute__((ext_vector_type(8))) float v8f;

#define RT2I  0.70710678118654752440f
#define MAGB  0.01f

// symmetric (half-sample) reflection into [0, l); valid for i >= -2l
__device__ __forceinline__ int reflect_idx(int i, int l) {
    int p = (i + 2 * l) % (2 * l);
    return (p < l) ? p : (2 * l - 1 - p);
}

__device__ __forceinline__ float magf(float re, float im) {
    return sqrtf(re * re + im * im + MAGB * MAGB) - MAGB;
}

// branchless banded-Toeplitz coefficient: h[k] if 0<=k<t else 0
__device__ __forceinline__ float tapval(const float* h, int k, int t) {
    int kc = k < 0 ? 0 : (k > t - 1 ? t - 1 : k);
    float v = h[kc];
    return (k == kc) ? v : 0.0f;
}

// ---------------------------------------------------------------------------
// Stage 1: fused rowfilter(h0o/h1o) + colfilter-as-WMMA + q2c + mag + avgpool.
// Input x: (96, 512, 512). Outputs: s01 (96,256,256), m1 (32,18,256,256).
// Workgroup: 128 threads = 4 waves, one 32x32 tile at 512-res.
// colfilter out[i] = sum_k h[k] * rowfilt[i + k - m], done as D = A(16x4)*B(4x16)+C
// over 6 K-slices covering a 24-row window (zero-padded Toeplitz band).
// ---------------------------------------------------------------------------
__global__ __launch_bounds__(128)
void k1_j1_fused(const float* __restrict__ x,
                 const float* __restrict__ h0o,
                 const float* __restrict__ h1o,
                 float* __restrict__ s01,
                 float* __restrict__ m1)
{
    __shared__ float s_in[38][40];
    __shared__ float s_lo[40][33];
    __shared__ float s_hi[40][33];
    __shared__ float s_q[4][32][33];   // ll, lh, hl, hh at 512-res (32x32 tile)
    __shared__ float s_h0[5];
    __shared__ float s_h1[7];

    const int tid = threadIdx.x;
    const int bc  = blockIdx.z;            // 0..95  (b*3 + ch)
    const int r0  = blockIdx.y * 32;
    const int c0  = blockIdx.x * 32;

    if (tid < 5) s_h0[tid] = h0o[tid];
    if (tid < 7) s_h1[tid] = h1o[tid];

    // load 38x38 halo tile (rows/cols r0-3 .. r0+34) with symmetric reflection
    const float* xin = x + (size_t)bc * (512 * 512);
    for (int p = tid; p < 38 * 38; p += 128) {
        int rr = p / 38, cc = p % 38;
        int gr = reflect_idx(r0 - 3 + rr, 512);
        int gc = reflect_idx(c0 - 3 + cc, 512);
        s_in[rr][cc] = xin[gr * 512 + gc];
    }
    __syncthreads();

    // rowfilter: lo = 5-tap h0o (m=2), hi = 7-tap h1o (m=3), 38 rows x 32 cols
    for (int p = tid; p < 38 * 32; p += 128) {
        int rr = p >> 5, c = p & 31;
        float lo = 0.f, hi = 0.f;
#pragma unroll
        for (int k = 0; k < 5; ++k) lo += s_h0[k] * s_in[rr][c + 1 + k];
#pragma unroll
        for (int k = 0; k < 7; ++k) hi += s_h1[k] * s_in[rr][c + k];
        s_lo[rr][c] = lo;
        s_hi[rr][c] = hi;
    }
    // rows 38,39 are read by the last K-slice with zero A-coeffs; keep finite
    if (tid < 64) {
        int rr = 38 + (tid >> 5), c = tid & 31;
        s_lo[rr][c] = 0.f;
        s_hi[rr][c] = 0.f;
    }
    __syncthreads();

    // ---- WMMA colfilter: wave -> one 16x16 quadrant -------------------------
    const int lane = tid & 31;
    const int wv   = tid >> 5;
    const int qr   = (wv >> 1) * 16;       // quadrant row base (0 or 16)
    const int qc   = (wv & 1) * 16;        // quadrant col base
    const int mi   = lane & 15;            // A: row M  /  B: col N
    const int kp0  = (lane < 16) ? 0 : 2;  // K components held by this half-wave

    v8f d_ll = {0.f,0.f,0.f,0.f,0.f,0.f,0.f,0.f};
    v8f d_lh = d_ll, d_hl = d_ll, d_hh = d_ll;

#pragma unroll
    for (int s = 0; s < 6; ++s) {
        const int p0 = 4 * s + kp0;        // window row offset for comp .x
        const int p1 = p0 + 1;             // and .y
        v2f a0, a1, blo, bhi;
        // A[M][p] = h[p - M - 3 + m] (zero outside band); h0: m=2, h1: m=3
        a0.x = tapval(s_h0, p0 - mi - 1, 5);
        a0.y = tapval(s_h0, p1 - mi - 1, 5);
        a1.x = tapval(s_h1, p0 - mi, 7);
        a1.y = tapval(s_h1, p1 - mi, 7);
        // B[p][N] = rowfiltered value at window row p, column N
        blo.x = s_lo[qr + p0][qc + mi];
        blo.y = s_lo[qr + p1][qc + mi];
        bhi.x = s_hi[qr + p0][qc + mi];
        bhi.y = s_hi[qr + p1][qc + mi];
        d_ll = __builtin_amdgcn_wmma_f32_16x16x4_f32(false, a0, false, blo, (short)0, d_ll, false, false);
        d_lh = __builtin_amdgcn_wmma_f32_16x16x4_f32(false, a1, false, blo, (short)0, d_lh, false, false);
        d_hl = __builtin_amdgcn_wmma_f32_16x16x4_f32(false, a0, false, bhi, (short)0, d_hl, false, false);
        d_hh = __builtin_amdgcn_wmma_f32_16x16x4_f32(false, a1, false, bhi, (short)0, d_hh, false, false);
    }

    // D layout: lane<16 -> M=v, N=lane ; lane>=16 -> M=v+8, N=lane-16
#pragma unroll
    for (int v = 0; v < 8; ++v) {
        int row = (lane < 16) ? v : v + 8;
        s_q[0][qr + row][qc + mi] = d_ll[v];
        s_q[1][qr + row][qc + mi] = d_lh[v];
        s_q[2][qr + row][qc + mi] = d_hl[v];
        s_q[3][qr + row][qc + mi] = d_hh[v];
    }
    __syncthreads();

    // ---- q2c + mag + avgpool: 16x16 outputs at 256-res ---------------------
    const int b = bc / 3, ch = bc % 3;
    const int I0 = r0 >> 1, J0 = c0 >> 1;
#pragma unroll
    for (int pp = 0; pp < 2; ++pp) {
        int p  = tid + pp * 128;
        int oi = p >> 4, oj = p & 15;
        int r = oi * 2, c = oj * 2;

        float la = s_q[0][r][c],   lb = s_q[0][r][c+1];
        float lc = s_q[0][r+1][c], ld = s_q[0][r+1][c+1];
        size_t sidx = ((size_t)bc * 256 + (I0 + oi)) * 256 + (J0 + oj);
        s01[sidx] = 0.25f * (la + lb + lc + ld);

        float va[3], vb[3], vc[3], vd[3];  // 0=lh, 1=hl, 2=hh
#pragma unroll
        for (int q = 0; q < 3; ++q) {
            va[q] = s_q[q+1][r][c]     * RT2I;
            vb[q] = s_q[q+1][r][c+1]   * RT2I;
            vc[q] = s_q[q+1][r+1][c]   * RT2I;
            vd[q] = s_q[q+1][r+1][c+1] * RT2I;
        }
        float mg[6];
        mg[0] = magf(va[0] - vd[0], vb[0] + vc[0]);   // 15  (lh)
        mg[1] = magf(va[2] - vd[2], vb[2] + vc[2]);   // 45  (hh)
        mg[2] = magf(va[1] - vd[1], vb[1] + vc[1]);   // 75  (hl)
        mg[3] = magf(va[1] + vd[1], vb[1] - vc[1]);   // 105 (hl)
        mg[4] = magf(va[2] + vd[2], vb[2] - vc[2]);   // 135 (hh)
        mg[5] = magf(va[0] + vd[0], vb[0] - vc[0]);   // 165 (lh)
#pragma unroll
        for (int o = 0; o < 6; ++o) {
            size_t midx = (((size_t)b * 18 + o * 3 + ch) * 256 + (I0 + oi)) * 256 + (J0 + oj);
            m1[midx] = mg[o];
        }
    }
}

// ---------------------------------------------------------------------------
// rowdfilt (stride-2 dual 10-tap, symmetric pad) on (C,256,256) -> lo/hi (C,256,128)
// lo = rowdfilt(x, h0b, h0a, lowpass) ; hi = rowdfilt(x, h1b, h1a, highpass)
// lo[2j]   = sum h0b[k] x[refl(4j+2k-8)] ; lo[2j+1] = sum h0a[k] x[refl(4j+2k-7)]
// hi[2j]   = sum h1a[k] x[refl(4j+2k-7)] ; hi[2j+1] = sum h1b[k] x[refl(4j+2k-8)]
// ---------------------------------------------------------------------------
__global__ __launch_bounds__(256)
void k_rowdfilt(const float* __restrict__ xin, int C,
                const float* __restrict__ h0a, const float* __restrict__ h0b,
                const float* __restrict__ h1a, const float* __restrict__ h1b,
                float* __restrict__ lo, float* __restrict__ hi)
{
    int idx = blockIdx.x * 256 + threadIdx.x;
    int total = C * 256 * 64;
    if (idx >= total) return;
    int j  = idx & 63;
    int r  = (idx >> 6) & 255;
    int ci = idx >> 14;
    const float* row = xin + ((size_t)ci * 256 + r) * 256;
    float lo0 = 0.f, lo1 = 0.f, hi0 = 0.f, hi1 = 0.f;
#pragma unroll
    for (int k = 0; k < 10; ++k) {
        float u = row[reflect_idx(4 * j + 2 * k - 8, 256)];
        float v = row[reflect_idx(4 * j + 2 * k - 7, 256)];
        lo0 += h0b[k] * u;
        lo1 += h0a[k] * v;
        hi0 += h1a[k] * v;
        hi1 += h1b[k] * u;
    }
    size_t o = ((size_t)ci * 256 + r) * 128 + 2 * j;
    lo[o] = lo0; lo[o + 1] = lo1;
    hi[o] = hi0; hi[o + 1] = hi1;
}

// ---------------------------------------------------------------------------
// Fused coldfilt + q2c + mag + 2x2-avg(ll) for one 64-res output pixel.
// lo/hi are (256,128) images. i,j in [0,64).
// ---------------------------------------------------------------------------
__device__ __forceinline__ void coldfilt_q2c(
    const float* __restrict__ lo, const float* __restrict__ hi,
    const float* __restrict__ h0a, const float* __restrict__ h0b,
    const float* __restrict__ h1a, const float* __restrict__ h1b,
    int i, int j, float mg[6], float* avg)
{
    const int c0 = 2 * j, c1 = c0 + 1;
    float ll_ya0=0.f, ll_ya1=0.f, ll_yb0=0.f, ll_yb1=0.f;
    float lh_ya0=0.f, lh_ya1=0.f, lh_yb0=0.f, lh_yb1=0.f;
    float hl_ya0=0.f, hl_ya1=0.f, hl_yb0=0.f, hl_yb1=0.f;
    float hh_ya0=0.f, hh_ya1=0.f, hh_yb0=0.f, hh_yb1=0.f;
#pragma unroll
    for (int k = 0; k < 10; ++k) {
        int ra = reflect_idx(4 * i + 2 * k - 8, 256);
        int rb = reflect_idx(4 * i + 2 * k - 7, 256);
        float u0 = lo[ra * 128 + c0], u1 = lo[ra * 128 + c1];
        float v0 = lo[rb * 128 + c0], v1 = lo[rb * 128 + c1];
        float w0 = hi[ra * 128 + c0], w1 = hi[ra * 128 + c1];
        float z0 = hi[rb * 128 + c0], z1 = hi[rb * 128 + c1];
        float f0b = h0b[k], f0a = h0a[k], f1b = h1b[k], f1a = h1a[k];
        ll_ya0 += f0b*u0; ll_ya1 += f0b*u1;
        ll_yb0 += f0a*v0; ll_yb1 += f0a*v1;
        lh_ya0 += f1b*u0; lh_ya1 += f1b*u1;
        lh_yb0 += f1a*v0; lh_yb1 += f1a*v1;
        hl_ya0 += f0b*w0; hl_ya1 += f0b*w1;
        hl_yb0 += f0a*z0; hl_yb1 += f0a*z1;
        hh_ya0 += f1b*w0; hh_ya1 += f1b*w1;
        hh_yb0 += f1a*z0; hh_yb1 += f1a*z1;
    }
    // ll (lowpass): row2i=ya, row2i+1=yb -> avgpool
    *avg = 0.25f * (ll_ya0 + ll_ya1 + ll_yb0 + ll_yb1);
    // q2c 2x2: a=row2i[c0], b=row2i[c1], c=row2i+1[c0], d=row2i+1[c1]
    float lha = lh_yb0*RT2I, lhb = lh_yb1*RT2I, lhc = lh_ya0*RT2I, lhd = lh_ya1*RT2I; // highpass order
    float hla = hl_ya0*RT2I, hlb = hl_ya1*RT2I, hlc = hl_yb0*RT2I, hld = hl_yb1*RT2I; // lowpass order
    float hha = hh_yb0*RT2I, hhb = hh_yb1*RT2I, hhc = hh_ya0*RT2I, hhd = hh_ya1*RT2I; // highpass order
    mg[0] = magf(lha - lhd, lhb + lhc);
    mg[1] = magf(hha - hhd, hhb + hhc);
    mg[2] = magf(hla - hld, hlb + hlc);
    mg[3] = magf(hla + hld, hlb - hlc);
    mg[4] = magf(hha + hhd, hhb - hhc);
    mg[5] = magf(lha + lhd, lhb - lhc);
}

// Stage 2: channels bc = b*3+ch; mags -> slots a=7..12, avg(ll) -> slot a=0
__global__ __launch_bounds__(256)
void k2_stage2_out(const float* __restrict__ lo, const float* __restrict__ hi,
                   const float* __restrict__ h0a, const float* __restrict__ h0b,
                   const float* __restrict__ h1a, const float* __restrict__ h1b,
                   float* __restrict__ out)
{
    int idx = blockIdx.x * 256 + threadIdx.x;
    if (idx >= 96 * 64 * 64) return;
    int j = idx & 63;
    int i = (idx >> 6) & 63;
    int bc = idx >> 12;
    int b = bc / 3, ch = bc % 3;
    const float* loimg = lo + (size_t)bc * (256 * 128);
    const float* hiimg = hi + (size_t)bc * (256 * 128);
    float mg[6], avg;
    coldfilt_q2c(loimg, hiimg, h0a, h0b, h1a, h1b, i, j, mg, &avg);
    size_t base = (size_t)b * 147;
    out[((base + ch) * 64 + i) * 64 + j] = avg;
#pragma unroll
    for (int o = 0; o < 6; ++o)
        out[((base + (size_t)(7 + o) * 3 + ch) * 64 + i) * 64 + j] = mg[o];
}

// Stage 3: channels cf = b*18 + o1*3 + ch; mags -> a=13+6*o2+o1, avg -> a=1+o1
__global__ __launch_bounds__(256)
void k3_stage3_out(const float* __restrict__ lo, const float* __restrict__ hi,
                   const float* __restrict__ h0a, const float* __restrict__ h0b,
                   const float* __restrict__ h1a, const float* __restrict__ h1b,
                   float* __restrict__ out)
{
    int idx = blockIdx.x * 256 + threadIdx.x;
    if (idx >= 576 * 64 * 64) return;
    int j = idx & 63;
    int i = (idx >> 6) & 63;
    int cf = idx >> 12;
    int b = cf / 18, t = cf % 18;
    int o1 = t / 3, ch = t % 3;
    const float* loimg = lo + (size_t)cf * (256 * 128);
    const float* hiimg = hi + (size_t)cf * (256 * 128);
    float mg[6], avg;
    coldfilt_q2c(loimg, hiimg, h0a, h0b, h1a, h1b, i, j, mg, &avg);
    size_t base = (size_t)b * 147;
    out[((base + (size_t)(1 + o1) * 3 + ch) * 64 + i) * 64 + j] = avg;
#pragma unroll
    for (int o2 = 0; o2 < 6; ++o2)
        out[((base + (size_t)(13 + o2 * 6 + o1) * 3 + ch) * 64 + i) * 64 + j] = mg[o2];
}

extern "C" void kernel_launch(void* const* d_in, const int* in_sizes, int n_in,
                              void* d_out, int out_size, void* d_ws, size_t ws_size,
                              hipStream_t stream)
{
    const float* x   = (const float*)d_in[0];
    const float* h0o = (const float*)d_in[1];
    const float* h1o = (const float*)d_in[2];
    const float* h0a = (const float*)d_in[3];
    const float* h0b = (const float*)d_in[4];
    const float* h1a = (const float*)d_in[5];
    const float* h1b = (const float*)d_in[6];
    float* out = (float*)d_out;

    char* ws = (char*)d_ws;
    const size_t S01_B = (size_t)96 * 256 * 256 * 4;        //  25.2 MB
    const size_t M1_B  = (size_t)32 * 18 * 256 * 256 * 4;   // 151.0 MB
    const size_t LH_B  = (size_t)576 * 256 * 128 * 4;       //  75.5 MB (sized for stage 3)
    float* s01 = (float*)(ws);
    float* m1  = (float*)(ws + S01_B);
    float* loB = (float*)(ws + S01_B + M1_B);
    float* hiB = (float*)(ws + S01_B + M1_B + LH_B);

    // Stage 1: j1 fused (rowfilter + WMMA colfilter + q2c/mag + avgpool)
    k1_j1_fused<<<dim3(16, 16, 96), 128, 0, stream>>>(x, h0o, h1o, s01, m1);

    // Stage 2: j2plus on s0_1 -> s1_j2 (a=7..12) and s0 (a=0)
    {
        int t1 = 96 * 256 * 64;
        k_rowdfilt<<<(t1 + 255) / 256, 256, 0, stream>>>(s01, 96, h0a, h0b, h1a, h1b, loB, hiB);
        int t2 = 96 * 64 * 64;
        k2_stage2_out<<<(t2 + 255) / 256, 256, 0, stream>>>(loB, hiB, h0a, h0b, h1a, h1b, out);
    }

    // Stage 3: j2plus on m1 -> s2 (a=13..48) and s1_j1 (a=1..6)
    {
        int t1 = 576 * 256 * 64;
        k_rowdfilt<<<(t1 + 255) / 256, 256, 0, stream>>>(m1, 576, h0a, h0b, h1a, h1b, loB, hiB);
        int t2 = 576 * 64 * 64;
        k3_stage3_out<<<(t2 + 255) / 256, 256, 0, stream>>>(loB, hiB, h0a, h0b, h1a, h1b, out);
    }
    (void)in_sizes; (void)n_in; (void)out_size; (void)ws_size;
}